// PASRNet_23914377904368
// MI455X (gfx1250) — compile-verified
//
#include <hip/hip_runtime.h>
#include <hip/hip_bf16.h>
#include <math.h>

// PASRNet forward for MI455X (gfx1250). All matmuls go through a single
// strided batched GEMM kernel built on V_WMMA_F32_16X16X4_F32 (fp32 matrix
// pipe -> numerics match the fp32 reference). Total ~60 GFLOP; working set
// fits in 192MB L2, so the GEMMs are compute-side, not HBM-bound.

#define BATCH 2
#define NPTS  1536
#define KNNK  20
#define LEAK  0.2f
#define BNEPS 1e-5f

typedef __attribute__((ext_vector_type(2))) float v2f;
typedef __attribute__((ext_vector_type(8))) float v8f;

#define F_BIAS  1
#define F_LEAKY 2
#define F_TANH  4

// ---------------------------------------------------------------------------
// Strided batched GEMM: D[b,m,n] = sum_k A[b,m,k]*B[b,k,n] (+bias[m]) (+act).
// One wave per 16x64 D tile; A fragment reused across 4 column tiles.
// WMMA fp32 fragment layouts per CDNA5 ISA 7.12.2:
//   A (16x4): lane l, comp v -> A[l&15, ((l>>4)<<1)+v]
//   B (4x16): lane l, comp v -> B[((l>>4)<<1)+v, l&15]
//   C/D     : lane l, comp v -> D[v + ((l>>4)<<3), l&15]
// Interior tiles (the overwhelming majority: every Nc here is a multiple of
// 64 and nearly every M of 16) take a branchless fast path with unconditional
// loads; edges use clamped addresses + value selects (v_cndmask, no exec
// branching, so EXEC stays all-ones around the WMMAs as the ISA requires).
// ---------------------------------------------------------------------------
__global__ __launch_bounds__(32) void k_gemm_wmma(
    const float* __restrict__ A, long Asm, long Ask, long Abs,
    const float* __restrict__ Bm, long Bsk, long Bsn, long Bbs,
    float* __restrict__ D, long Dld, long Dbs,
    const float* __restrict__ bias,
    int M, int Nc, int K, int flags)
{
  const int lane = threadIdx.x & 31;
  const int tm = blockIdx.y << 4;
  const int tn = blockIdx.x << 6;
  A  += (long)blockIdx.z * Abs;
  Bm += (long)blockIdx.z * Bbs;
  D  += (long)blockIdx.z * Dbs;

  const int row  = lane & 15;        // A row / B col / D col (within 16)
  const int koff = (lane >> 4) << 1; // 0 or 2

  v8f acc[4] = {{}, {}, {}, {}};

  const bool interior = (tm + 16 <= M) && (tn + 64 <= Nc);
  const int K4 = K & ~3;

  if (interior) {
    // ---- fast path: unconditional loads, strength-reduced addresses ----
    const float* pA  = A  + (long)(tm + row) * Asm + (long)koff * Ask;
    const float* pB0 = Bm + (long)koff * Bsk + (long)(tn + row) * Bsn;
    const float* pB1 = pB0 + 16 * Bsn;
    const float* pB2 = pB1 + 16 * Bsn;
    const float* pB3 = pB2 + 16 * Bsn;
    const long aInc = 4 * Ask, bInc = 4 * Bsk;
#pragma unroll 2
    for (int kk = 0; kk < K4; kk += 4) {
      v2f a, b0, b1, b2, b3;
      a.x  = pA[0];   a.y  = pA[Ask];
      b0.x = pB0[0];  b0.y = pB0[Bsk];
      b1.x = pB1[0];  b1.y = pB1[Bsk];
      b2.x = pB2[0];  b2.y = pB2[Bsk];
      b3.x = pB3[0];  b3.y = pB3[Bsk];
      acc[0] = __builtin_amdgcn_wmma_f32_16x16x4_f32(false, a, false, b0, (short)0, acc[0], false, false);
      acc[1] = __builtin_amdgcn_wmma_f32_16x16x4_f32(false, a, false, b1, (short)0, acc[1], false, false);
      acc[2] = __builtin_amdgcn_wmma_f32_16x16x4_f32(false, a, false, b2, (short)0, acc[2], false, false);
      acc[3] = __builtin_amdgcn_wmma_f32_16x16x4_f32(false, a, false, b3, (short)0, acc[3], false, false);
      pA += aInc; pB0 += bInc; pB1 += bInc; pB2 += bInc; pB3 += bInc;
    }
    if (K4 < K) {  // K remainder: clamp k, select loaded values (branchless)
      const int k0 = K4 + koff;
      const bool ok0 = k0 < K, ok1 = (k0 + 1) < K;
      const long ka = ok0 ? k0 : 0, kb = ok1 ? (k0 + 1) : 0;
      const float* Ar = A + (long)(tm + row) * Asm;
      float a0 = Ar[ka * Ask], a1 = Ar[kb * Ask];
      v2f a; a.x = ok0 ? a0 : 0.f; a.y = ok1 ? a1 : 0.f;
#pragma unroll
      for (int j = 0; j < 4; ++j) {
        const long n = tn + (j << 4) + row;
        float u0 = Bm[ka * Bsk + n * Bsn];
        float u1 = Bm[kb * Bsk + n * Bsn];
        v2f bv; bv.x = ok0 ? u0 : 0.f; bv.y = ok1 ? u1 : 0.f;
        acc[j] = __builtin_amdgcn_wmma_f32_16x16x4_f32(false, a, false, bv, (short)0, acc[j], false, false);
      }
    }
  } else {
    // ---- edge path: clamped addresses + value selects, still branchless ----
    const int m  = tm + row;
    const long mc = (m < M) ? m : 0;
    const bool mok = m < M;
    for (int kk = 0; kk < K; kk += 4) {
      const int k0 = kk + koff;
      const bool ok0 = k0 < K, ok1 = (k0 + 1) < K;
      const long ka = ok0 ? k0 : 0, kb = ok1 ? (k0 + 1) : 0;
      float a0 = A[mc * Asm + ka * Ask];
      float a1 = A[mc * Asm + kb * Ask];
      v2f a;
      a.x = (mok && ok0) ? a0 : 0.f;
      a.y = (mok && ok1) ? a1 : 0.f;
#pragma unroll
      for (int j = 0; j < 4; ++j) {
        const int n = tn + (j << 4) + row;
        const long nc = (n < Nc) ? n : 0;
        const bool nok = n < Nc;
        float u0 = Bm[ka * Bsk + nc * Bsn];
        float u1 = Bm[kb * Bsk + nc * Bsn];
        v2f bv;
        bv.x = (nok && ok0) ? u0 : 0.f;
        bv.y = (nok && ok1) ? u1 : 0.f;
        acc[j] = __builtin_amdgcn_wmma_f32_16x16x4_f32(false, a, false, bv, (short)0, acc[j], false, false);
      }
    }
  }

  // ---- epilogue / store ----
  const int dm0 = tm + ((lane >> 4) << 3);
  if (interior) {
#pragma unroll
    for (int j = 0; j < 4; ++j) {
      const long n = tn + (j << 4) + row;
#pragma unroll
      for (int v = 0; v < 8; ++v) {
        const long m = dm0 + v;
        float val = acc[j][v];
        if (flags & F_BIAS)  val += bias[m];
        if (flags & F_LEAKY) val = (val > 0.f) ? val : LEAK * val;
        if (flags & F_TANH)  val = 0.5f * tanhf(val);
        D[m * Dld + n] = val;
      }
    }
  } else {
#pragma unroll
    for (int j = 0; j < 4; ++j) {
      const int n = tn + (j << 4) + row;
      if (n >= Nc) continue;
#pragma unroll
      for (int v = 0; v < 8; ++v) {
        const int m = dm0 + v;
        if (m >= M) continue;
        float val = acc[j][v];
        if (flags & F_BIAS)  val += bias[m];
        if (flags & F_LEAKY) val = (val > 0.f) ? val : LEAK * val;
        if (flags & F_TANH)  val = 0.5f * tanhf(val);
        D[(long)m * Dld + n] = val;
      }
    }
  }
}

// ---------------------------------------------------------------------------
// pc (B,N,3) -> x (B,3,N)
// ---------------------------------------------------------------------------
__global__ void k_transpose_pc(const float* __restrict__ pc, float* __restrict__ x)
{
  long i = (long)blockIdx.x * blockDim.x + threadIdx.x;
  const long tot = (long)BATCH * NPTS * 3;
  if (i >= tot) return;
  int c = (int)(i % 3);
  long t = i / 3;
  int n = (int)(t % NPTS);
  int b = (int)(t / NPTS);
  x[((long)b * 3 + c) * NPTS + n] = pc[i];
}

// ---------------------------------------------------------------------------
// kNN: per (b,n) compute pd[m] = 2<xn,xm> - |xn|^2 - |xm|^2 into LDS, then
// 20 iterative argmax passes (matches top_k of -squared-distance; only the
// index *set* matters since a max over K follows).
// ---------------------------------------------------------------------------
#define KNT 128
__global__ __launch_bounds__(KNT) void k_knn(
    const float* __restrict__ x, int C, long xbs, int* __restrict__ idx_out)
{
  __shared__ float pd[NPTS];
  __shared__ float rv[KNT];
  __shared__ int   ri[KNT];
  const int n = blockIdx.x;
  const int b = blockIdx.y;
  const float* xb = x + (long)b * xbs;
  const int tid = threadIdx.x;

  float xxn = 0.f;
  for (int c = 0; c < C; ++c) { float v = xb[(long)c * NPTS + n]; xxn += v * v; }

  for (int m = tid; m < NPTS; m += KNT) {
    float inner = 0.f, xxm = 0.f;
    for (int c = 0; c < C; ++c) {
      float xm = xb[(long)c * NPTS + m];
      float xn = xb[(long)c * NPTS + n];
      inner += xm * xn;
      xxm   += xm * xm;
    }
    pd[m] = 2.f * inner - xxn - xxm;
  }
  __syncthreads();

  for (int k = 0; k < KNNK; ++k) {
    float best = -3.4e38f; int bi = 0;
    for (int m = tid; m < NPTS; m += KNT)
      if (pd[m] > best) { best = pd[m]; bi = m; }
    rv[tid] = best; ri[tid] = bi;
    __syncthreads();
    for (int s = KNT / 2; s > 0; s >>= 1) {
      if (tid < s && rv[tid + s] > rv[tid]) { rv[tid] = rv[tid + s]; ri[tid] = ri[tid + s]; }
      __syncthreads();
    }
    if (tid == 0) {
      int w = ri[0];
      idx_out[((long)b * NPTS + n) * KNNK + k] = w;
      pd[w] = -3.4e38f;
    }
    __syncthreads();
  }
}

// ---------------------------------------------------------------------------
// g (B,2C,N,K): c<C -> x[b,c,idx]-x[b,c,n] ; c>=C -> x[b,c-C,n]
// ---------------------------------------------------------------------------
__global__ void k_build_graph(const float* __restrict__ x, long xbs,
                              const int* __restrict__ idx,
                              float* __restrict__ g, int C)
{
  long i = (long)blockIdx.x * blockDim.x + threadIdx.x;
  const long tot = (long)BATCH * 2 * C * NPTS * KNNK;
  if (i >= tot) return;
  int k = (int)(i % KNNK);
  long t = i / KNNK;
  int n = (int)(t % NPTS); t /= NPTS;
  int c = (int)(t % (2 * C));
  int b = (int)(t / (2 * C));
  float val;
  if (c < C) {
    int j = idx[((long)b * NPTS + n) * KNNK + k];
    val = x[(long)b * xbs + (long)c * NPTS + j] - x[(long)b * xbs + (long)c * NPTS + n];
  } else {
    val = x[(long)b * xbs + (long)(c - C) * NPTS + n];
  }
  g[i] = val;
}

// ---------------------------------------------------------------------------
// out[b, coff+c, n] = max_k h[b,c,n,k]     (h: B x C x N x K contiguous)
// ---------------------------------------------------------------------------
__global__ void k_max_k(const float* __restrict__ h, float* __restrict__ outp,
                        long obs, int coff, int C)
{
  long i = (long)blockIdx.x * blockDim.x + threadIdx.x;
  const long tot = (long)BATCH * C * NPTS;
  if (i >= tot) return;
  int n = (int)(i % NPTS);
  long t = i / NPTS;
  int c = (int)(t % C);
  int b = (int)(t / C);
  const float* p = h + (((long)(b * C + c) * NPTS) + n) * KNNK;
  float m = p[0];
  for (int k = 1; k < KNNK; ++k) m = fmaxf(m, p[k]);
  outp[(long)b * obs + (long)(coff + c) * NPTS + n] = m;
}

// ---------------------------------------------------------------------------
// BN stats: per channel over (B, N): scale = g*rsqrt(var+eps), shift = be-m*scale
// ---------------------------------------------------------------------------
__global__ __launch_bounds__(256) void k_bn_stats(
    const float* __restrict__ xin, long cbs, int C,
    const float* __restrict__ gamma, const float* __restrict__ beta,
    float* __restrict__ scale, float* __restrict__ shift)
{
  const int c = blockIdx.x;
  __shared__ float s1[256], s2[256];
  float a = 0.f, q = 0.f;
  const int tot = BATCH * NPTS;
  for (int i = threadIdx.x; i < tot; i += 256) {
    int bb = i / NPTS, n = i % NPTS;
    float v = xin[(long)bb * cbs + (long)c * NPTS + n];
    a += v; q += v * v;
  }
  s1[threadIdx.x] = a; s2[threadIdx.x] = q;
  __syncthreads();
  for (int s = 128; s > 0; s >>= 1) {
    if (threadIdx.x < s) { s1[threadIdx.x] += s1[threadIdx.x + s]; s2[threadIdx.x] += s2[threadIdx.x + s]; }
    __syncthreads();
  }
  if (threadIdx.x == 0) {
    float m   = s1[0] / (float)tot;
    float var = s2[0] / (float)tot - m * m;
    float sc  = gamma[c] * rsqrtf(var + BNEPS);
    scale[c] = sc;
    shift[c] = beta[c] - m * sc;
  }
}

__global__ void k_bn_apply(const float* __restrict__ xin, long inbs,
                           float* __restrict__ outp, long outbs,
                           const float* __restrict__ scale,
                           const float* __restrict__ shift, int C, int relu)
{
  long i = (long)blockIdx.x * blockDim.x + threadIdx.x;
  const long tot = (long)BATCH * C * NPTS;
  if (i >= tot) return;
  int n = (int)(i % NPTS);
  long t = i / NPTS;
  int c = (int)(t % C);
  int b = (int)(t / C);
  float v = xin[(long)b * inbs + (long)c * NPTS + n] * scale[c] + shift[c];
  if (relu) v = fmaxf(v, 0.f);
  outp[(long)b * outbs + (long)c * NPTS + n] = v;
}

// glob[b*1024+c] = max_n y[b,c,n]   (y: B x 1024 x N contiguous)
__global__ __launch_bounds__(256) void k_global_max(
    const float* __restrict__ y, float* __restrict__ glob)
{
  const int bc = blockIdx.x;
  __shared__ float sm[256];
  const float* p = y + (long)bc * NPTS;
  float m = -3.4e38f;
  for (int n = threadIdx.x; n < NPTS; n += 256) m = fmaxf(m, p[n]);
  sm[threadIdx.x] = m;
  __syncthreads();
  for (int s = 128; s > 0; s >>= 1) {
    if (threadIdx.x < s) sm[threadIdx.x] = fmaxf(sm[threadIdx.x], sm[threadIdx.x + s]);
    __syncthreads();
  }
  if (threadIdx.x == 0) glob[bc] = sm[0];
}

// z[b, C+c, n] = glob[b*C+c]  (z has zC channels)
__global__ void k_broadcast(const float* __restrict__ glob, float* __restrict__ z,
                            int C, int zC)
{
  long i = (long)blockIdx.x * blockDim.x + threadIdx.x;
  const long tot = (long)BATCH * C * NPTS;
  if (i >= tot) return;
  int n = (int)(i % NPTS);
  long t = i / NPTS;
  int c = (int)(t % C);
  int b = (int)(t / C);
  z[((long)b * zC + C + c) * NPTS + n] = glob[b * C + c];
}

// ---------------------------------------------------------------------------
extern "C" void kernel_launch(void* const* d_in, const int* in_sizes, int n_in,
                              void* d_out, int out_size, void* d_ws, size_t ws_size,
                              hipStream_t stream)
{
  (void)in_sizes; (void)n_in; (void)out_size; (void)ws_size;
  const float* pc     = (const float*)d_in[0];
  const float* adj    = (const float*)d_in[1];
  const float* fe_w1a = (const float*)d_in[2];
  const float* fe_w1b = (const float*)d_in[3];
  const float* fe_w2a = (const float*)d_in[4];
  const float* fe_w2b = (const float*)d_in[5];
  const float* gcn_w[4], *gcn_b[4], *gcn_g[4], *gcn_be[4];
  for (int i = 0; i < 4; ++i) {
    gcn_w[i]  = (const float*)d_in[6 + 4 * i];
    gcn_b[i]  = (const float*)d_in[7 + 4 * i];
    gcn_g[i]  = (const float*)d_in[8 + 4 * i];
    gcn_be[i] = (const float*)d_in[9 + 4 * i];
  }
  const float* pn_w[3], *pn_b[3], *pn_g[3], *pn_be[3];
  for (int i = 0; i < 3; ++i) {
    pn_w[i]  = (const float*)d_in[22 + 4 * i];
    pn_b[i]  = (const float*)d_in[23 + 4 * i];
    pn_g[i]  = (const float*)d_in[24 + 4 * i];
    pn_be[i] = (const float*)d_in[25 + 4 * i];
  }
  const float* dec_w[3], *dec_b[3], *dec_g[3], *dec_be[3];
  for (int i = 0; i < 3; ++i) {
    dec_w[i]  = (const float*)d_in[34 + 4 * i];
    dec_b[i]  = (const float*)d_in[35 + 4 * i];
    dec_g[i]  = (const float*)d_in[36 + 4 * i];
    dec_be[i] = (const float*)d_in[37 + 4 * i];
  }
  const float* dec_w4 = (const float*)d_in[46];
  const float* dec_b4 = (const float*)d_in[47];
  float* out = (float*)d_out;

  const long N  = NPTS;
  const long NK = (long)NPTS * KNNK;

  // workspace layout (floats, 256-aligned)
  float* cur = (float*)d_ws;
  auto take = [&](long nf) { float* p = cur; cur += ((nf + 255) & ~255L); return p; };
  float* x    = take((long)BATCH * 3 * N);
  int*   idx  = (int*)take((long)BATCH * N * KNNK);
  float* g    = take((long)BATCH * 128 * NK);
  float* eh0  = take((long)BATCH * 64 * NK);
  float* eh1  = take((long)BATCH * 64 * NK);
  float* xper = take((long)BATCH * 128 * N);
  float* glob = take((long)BATCH * 1024);
  float* scl  = take(2048);
  float* sft  = take(2048);
  float* big0 = take((long)BATCH * 2048 * N);  // scratch / decoder wide
  float* big1 = take((long)BATCH * 1024 * N);  // GCN h / decoder narrow
  float* big2 = take((long)BATCH * 2048 * N);  // z = [gcn_h ; glob]

  auto gemm = [&](const float* Ap, long Asm_, long Ask_, long Abs_,
                  const float* Bp, long Bsk_, long Bsn_, long Bbs_,
                  float* Dp, long Dld_, long Dbs_,
                  const float* bias, int M, int Ncc, int Kd, int flags) {
    dim3 grid((Ncc + 63) / 64, (M + 15) / 16, BATCH);
    k_gemm_wmma<<<grid, dim3(32), 0, stream>>>(
        Ap, Asm_, Ask_, Abs_, Bp, Bsk_, Bsn_, Bbs_, Dp, Dld_, Dbs_,
        bias, M, Ncc, Kd, flags);
  };
  auto bn = [&](const float* in, long inbs, float* op, long outbs, int C,
                const float* gam, const float* bet, int relu) {
    k_bn_stats<<<C, 256, 0, stream>>>(in, inbs, C, gam, bet, scl, sft);
    long tot = (long)BATCH * C * N;
    k_bn_apply<<<(unsigned)((tot + 255) / 256), 256, 0, stream>>>(
        in, inbs, op, outbs, scl, sft, C, relu);
  };
  auto blocks = [](long tot) { return (unsigned)((tot + 255) / 256); };

  // ---- x = pc^T --------------------------------------------------------
  k_transpose_pc<<<blocks((long)BATCH * N * 3), 256, 0, stream>>>(pc, x);

  // ---- EdgeConv block 1 (C=3) -----------------------------------------
  k_knn<<<dim3(NPTS, BATCH), KNT, 0, stream>>>(x, 3, 3 * N, idx);
  k_build_graph<<<blocks((long)BATCH * 6 * NK), 256, 0, stream>>>(x, 3 * N, idx, g, 3);
  gemm(fe_w1a, 6, 1, 0,   g,   NK, 1, 6 * NK,  eh0, NK, 64 * NK, nullptr, 64, (int)NK, 6,  F_LEAKY);
  gemm(fe_w1b, 64, 1, 0,  eh0, NK, 1, 64 * NK, eh1, NK, 64 * NK, nullptr, 64, (int)NK, 64, F_LEAKY);
  k_max_k<<<blocks((long)BATCH * 64 * N), 256, 0, stream>>>(eh1, xper, 128 * N, 0, 64);

  // ---- EdgeConv block 2 (C=64; x1 = xper[:, :64]) ---------------------
  k_knn<<<dim3(NPTS, BATCH), KNT, 0, stream>>>(xper, 64, 128 * N, idx);
  k_build_graph<<<blocks((long)BATCH * 128 * NK), 256, 0, stream>>>(xper, 128 * N, idx, g, 64);
  gemm(fe_w2a, 128, 1, 0, g,   NK, 1, 128 * NK, eh0, NK, 64 * NK, nullptr, 64, (int)NK, 128, F_LEAKY);
  gemm(fe_w2b, 64, 1, 0,  eh0, NK, 1, 64 * NK,  eh1, NK, 64 * NK, nullptr, 64, (int)NK, 64,  F_LEAKY);
  k_max_k<<<blocks((long)BATCH * 64 * N), 256, 0, stream>>>(eh1, xper, 128 * N, 64, 64);

  // ---- GCN stack: h = relu(bn(W (h @ adj) + b)) -----------------------
  const int cin[4]  = {128, 512, 512, 1024};
  const int cout[4] = {512, 512, 1024, 1024};
  const float* h = xper;
  long hbs = 128 * N;
  for (int i = 0; i < 4; ++i) {
    // t = h @ adj : A[m=c,k=n]=h (Asm=N,Ask=1), B[k=n,n=m]=adj (Bsk=N,Bsn=1)
    gemm(h, N, 1, hbs, adj, N, 1, N * N, big0, N, (long)cin[i] * N,
         nullptr, cin[i], (int)N, (int)N, 0);
    // u = W t + b
    gemm(gcn_w[i], cin[i], 1, 0, big0, N, 1, (long)cin[i] * N,
         big1, N, (long)cout[i] * N, gcn_b[i], cout[i], (int)N, cin[i], F_BIAS);
    if (i < 3)
      bn(big1, (long)cout[i] * N, big1, (long)cout[i] * N, cout[i], gcn_g[i], gcn_be[i], 1);
    else  // final layer: write straight into z's first 1024 channels
      bn(big1, (long)cout[i] * N, big2, 2048 * N, cout[i], gcn_g[i], gcn_be[i], 1);
    h = big1; hbs = (long)cout[i] * N;
  }

  // ---- PointNet branch -------------------------------------------------
  gemm(pn_w[0], 3, 1, 0,   x,    N, 1, 3 * N,   big0, N, 64 * N,   pn_b[0], 64,   (int)N, 3,   F_BIAS);
  bn(big0, 64 * N, big0, 64 * N, 64, pn_g[0], pn_be[0], 1);
  gemm(pn_w[1], 64, 1, 0,  big0, N, 1, 64 * N,  big1, N, 128 * N,  pn_b[1], 128,  (int)N, 64,  F_BIAS);
  bn(big1, 128 * N, big1, 128 * N, 128, pn_g[1], pn_be[1], 1);
  gemm(pn_w[2], 128, 1, 0, big1, N, 1, 128 * N, big0, N, 1024 * N, pn_b[2], 1024, (int)N, 128, F_BIAS);
  bn(big0, 1024 * N, big0, 1024 * N, 1024, pn_g[2], pn_be[2], 0);  // no relu
  k_global_max<<<BATCH * 1024, 256, 0, stream>>>(big0, glob);
  k_broadcast<<<blocks((long)BATCH * 1024 * N), 256, 0, stream>>>(glob, big2, 1024, 2048);

  // ---- Decoder ---------------------------------------------------------
  gemm(dec_w[0], 2048, 1, 0, big2, N, 1, 2048 * N, big0, N, 2048 * N,
       dec_b[0], 2048, (int)N, 2048, F_BIAS);
  bn(big0, 2048 * N, big0, 2048 * N, 2048, dec_g[0], dec_be[0], 1);
  gemm(dec_w[1], 2048, 1, 0, big0, N, 1, 2048 * N, big1, N, 512 * N,
       dec_b[1], 512, (int)N, 2048, F_BIAS);
  bn(big1, 512 * N, big1, 512 * N, 512, dec_g[1], dec_be[1], 1);
  gemm(dec_w[2], 512, 1, 0, big1, N, 1, 512 * N, big0, N, 256 * N,
       dec_b[2], 256, (int)N, 512, F_BIAS);
  bn(big0, 256 * N, big0, 256 * N, 256, dec_g[2], dec_be[2], 1);
  // final: out = 0.5*tanh(W4 d3 + b4), written directly to d_out (B,3,N)
  gemm(dec_w4, 256, 1, 0, big0, N, 1, 256 * N, out, N, 3 * N,
       dec_b4, 3, (int)N, 256, F_BIAS | F_TANH);
}